// Deep_Tensor_Net_11441792877146
// MI455X (gfx1250) — compile-verified
//
#include <hip/hip_runtime.h>

// ---------------------------------------------------------------------------
// out[B,384]:  out[q,i] = sum_{a,b} A[a,i,b] * G_q[b,a]
//              G_q[b,a] = sum_c Bf[b, x[q,0], c] * C[c, x[q,1], a]
// => GEMM  out = G[B x 64] * Aflat^T[64 x 384]  via V_WMMA_F32_16X16X4_F32
//    (k = a*8 + b ;  Aflat[i][k] = A[a,i,b] ;  g_q[k] = G_q[b,a])
// Roofline: 25.8 GFLOP vs 0.81 GB mandatory traffic -> ~35us HBM floor.
// fp32 WMMA keeps exact reference precision; 16x16x4 k-steps chained x16.
// ---------------------------------------------------------------------------

typedef float v2f __attribute__((ext_vector_type(2)));
typedef float v4f __attribute__((ext_vector_type(4)));
typedef float v8f __attribute__((ext_vector_type(8)));

#define N_DIM 384
#define R_DIM 8
#define PAIR_ROW 400           // padded float2 per kh-row (384 used); 3200B pitch
                               //  -> +row = +800 dwords = +32 banks: halves of the
                               //     wave hit disjoint 32-bank groups (conflict-free b64)
#define QT_PER_BLOCK 128       // 8 waves x 16 queries
#define LDS_BYTES (32 * PAIR_ROW * sizeof(float2))   // 102400 B  (<= 320KB WGP LDS)

__global__ __launch_bounds__(256)
void dtn_tr_gemm_kernel(const float* __restrict__ A,
                        const float* __restrict__ Bf,
                        const float* __restrict__ C,
                        const int*   __restrict__ x,
                        float* __restrict__ out)
{
    extern __shared__ float2 lds[];   // lds[kh*PAIR_ROW + n] = {Aflat[n][2kh], Aflat[n][2kh+1]}

    const int tid = threadIdx.x;

    // ---- Stage Aflat into LDS in WMMA B-operand pair layout.
    //      k = a*8 + b ; pair (2kh, 2kh+1) shares 'a' and has consecutive 'b'
    //      => contiguous 8-byte global load from A[a, n, b..b+1].
    for (int idx = tid; idx < 32 * N_DIM; idx += 256) {
        int kh = idx / N_DIM;
        int n  = idx - kh * N_DIM;
        int a  = kh >> 2;
        int b2 = (kh & 3) * 2;
        const float* src = A + ((size_t)(a * N_DIM + n) * R_DIM + b2);
        lds[kh * PAIR_ROW + n] = make_float2(src[0], src[1]);
    }
    __syncthreads();

    const int wave = tid >> 5;
    const int lane = tid & 31;
    const int hi   = lane >> 4;     // lanes 16-31 carry K = 2,3 (mod 4) of the A-operand
    const int lq   = lane & 15;     // query within the wave's 16-query M-tile
    const int qBase = blockIdx.x * QT_PER_BLOCK + wave * 16;
    const int q     = qBase + lq;

    const int jq = x[2 * q + 0];
    const int kq = x[2 * q + 1];

    // ---- Gather the 8x8 slices needed for G (b128 loads; tables are cache-resident).
    v4f cv[8][2];                        // cv[c][.] = C[c, kq, 0..7]
    #pragma unroll
    for (int c = 0; c < 8; ++c) {
        const float* p = C + ((size_t)(c * N_DIM + kq) * R_DIM);
        cv[c][0] = *(const v4f*)(p);
        cv[c][1] = *(const v4f*)(p + 4);
    }
    v4f bvv[4][2];                       // bvv[m][.] = Bf[b2(m), jq, 0..7]
    #pragma unroll
    for (int m = 0; m < 4; ++m) {
        int b2 = 2 * hi + (m & 1) + 4 * (m >> 1);
        const float* p = Bf + ((size_t)(b2 * N_DIM + jq) * R_DIM);
        bvv[m][0] = *(const v4f*)(p);
        bvv[m][1] = *(const v4f*)(p + 4);
    }

    // ---- Per-lane A-operand values, pre-arranged per WMMA k-step:
    //      g[kk] = { g_q[4kk + 2hi + 0], g_q[4kk + 2hi + 1] }
    v2f g[16];
    #pragma unroll
    for (int kk = 0; kk < 16; ++kk) {
        const int a  = kk >> 1;
        const int m0 = 2 * (kk & 1);
        float s0 = 0.f, s1 = 0.f;
        #pragma unroll
        for (int c = 0; c < 8; ++c) {
            float cc = cv[c][a >> 2][a & 3];
            s0 += bvv[m0 + 0][c >> 2][c & 3] * cc;
            s1 += bvv[m0 + 1][c >> 2][c & 3] * cc;
        }
        g[kk].x = s0;
        g[kk].y = s1;
    }

    // ---- GEMM sweep across the 24 N-tiles: 16 chained fp32 WMMAs per tile.
    float* outBase = out + (size_t)qBase * N_DIM;
    #pragma unroll 2
    for (int nt = 0; nt < 24; ++nt) {
        const int n0 = nt * 16;
        v8f acc = {0.f, 0.f, 0.f, 0.f, 0.f, 0.f, 0.f, 0.f};
        #pragma unroll
        for (int kk = 0; kk < 16; ++kk) {
            // B-operand: lane = N, VGPR0/1 = K = 4kk + 2hi + {0,1}  (conflict-free b64)
            float2 bp = lds[(2 * kk + hi) * PAIR_ROW + n0 + lq];
            v2f bop;
            bop.x = bp.x;
            bop.y = bp.y;
            acc = __builtin_amdgcn_wmma_f32_16x16x4_f32(
                false, g[kk], false, bop, (short)0, acc, false, false);
        }
        // D layout: VGPR r -> M = r + 8*hi, lane -> N ; 16-lane rows = 64B segments
        float* o = outBase + (size_t)(8 * hi) * N_DIM + n0 + lq;
        #pragma unroll
        for (int r = 0; r < 8; ++r)
            o[(size_t)r * N_DIM] = acc[r];
    }
}

extern "C" void kernel_launch(void* const* d_in, const int* in_sizes, int n_in,
                              void* d_out, int out_size, void* d_ws, size_t ws_size,
                              hipStream_t stream) {
    const float* A  = (const float*)d_in[0];
    const float* Bf = (const float*)d_in[1];
    const float* C  = (const float*)d_in[2];
    const int*   x  = (const int*)d_in[3];
    float* out = (float*)d_out;

    const int B = in_sizes[3] / 2;              // 524288
    const int blocks = B / QT_PER_BLOCK;        // 4096

    // 100KB dynamic LDS (> default cap) — raise the limit; deterministic host call.
    hipFuncSetAttribute((const void*)dtn_tr_gemm_kernel,
                        hipFuncAttributeMaxDynamicSharedMemorySize, (int)LDS_BYTES);

    dtn_tr_gemm_kernel<<<blocks, 256, LDS_BYTES, stream>>>(A, Bf, C, x, out);
}